// CARD_9165460209973
// MI455X (gfx1250) — compile-verified
//
#include <hip/hip_runtime.h>
#include <math.h>

// ---------------- problem constants ----------------
#define B_     4
#define S_     720
#define CIN_   321
#define PATCH_ 16
#define STRIDE_ 8
#define D_     128
#define NH_    8
#define DFF_   256
#define DP_    8
#define EL_    2
#define PRED_  96
#define HD_    16
#define PN_    89
#define TTN_   90
#define EMA_   321

typedef float v2f __attribute__((ext_vector_type(2)));
typedef float v8f __attribute__((ext_vector_type(8)));

// ---------------- generic batched fp32 WMMA GEMM ----------------
// C[b0,b1,b2][M,N] = alpha * A x B + bias[n] + Add ; optional exact GELU.
// transA: A stored K-major (A[k*ldA+m]); transB: B stored N-major (B[n*ldB+k]).
struct GemmDesc {
  const float* A; const float* Bm; const float* bias; const float* Add; float* C;
  long ldA, sA0, sA1, sA2;
  long ldB, sB0, sB1, sB2;
  long ldAdd, aS0, aS1, aS2;
  long ldC, sC0, sC1, sC2;
  int M, N, K;
  int b0, b1, b2;
  float alpha;
  int transA, transB, gelu;
  int tm, tn; long totalTiles;
};

__device__ __forceinline__ float ep_apply(const GemmDesc& g, float v, int nn,
                                          const float* Add, long m) {
  v *= g.alpha;
  if (g.bias) v += g.bias[nn];
  if (Add)    v += Add[m*g.ldAdd + nn];
  if (g.gelu) v = 0.5f * v * (1.0f + erff(v * 0.70710678118654752f));
  return v;
}

// ---- 16x16 tile per wave (used for small-N GEMMs) ----
__global__ __launch_bounds__(256) void wmma_gemm_f32(GemmDesc g) {
  long wave = (long)blockIdx.x * 8 + (threadIdx.x >> 5);
  if (wave >= g.totalTiles) return;
  int lane = threadIdx.x & 31;
  int half = lane >> 4, ln = lane & 15;
  long t = wave;
  int jn = (int)(t % g.tn); t /= g.tn;
  int im = (int)(t % g.tm); t /= g.tm;
  int i2 = (int)(t % g.b2); t /= g.b2;
  int i1 = (int)(t % g.b1); t /= g.b1;
  int i0 = (int)t;
  const float* A  = g.A  + (long)i0*g.sA0 + (long)i1*g.sA1 + (long)i2*g.sA2;
  const float* Bp = g.Bm + (long)i0*g.sB0 + (long)i1*g.sB1 + (long)i2*g.sB2;
  int m0 = im*16, n0 = jn*16;
  int bn = n0 + ln;                       // B/C column this lane owns
  bool bnv = bn < g.N;
  // Clamp load row/col in-bounds: garbage only lands in C rows/cols we mask at store.
  int am_c = m0 + ln; if (am_c > g.M - 1) am_c = g.M - 1;
  int bn_c = bn;      if (bn_c > g.N - 1) bn_c = g.N - 1;
  const long aE = g.transA ? g.ldA : 1;
  const long bE = g.transB ? 1 : g.ldB;
  const float* pA = (g.transA ? (A + am_c) : (A + (long)am_c*g.ldA)) + (long)(2*half)*aE;
  const float* pB = (g.transB ? (Bp + (long)bn_c*g.ldB) : (Bp + bn_c)) + (long)(2*half)*bE;

  v8f acc = {0.f,0.f,0.f,0.f,0.f,0.f,0.f,0.f};
  const int K4 = g.K & ~3;
  #pragma unroll 2
  for (int k0 = 0; k0 < K4; k0 += 4) {
    v2f av = { pA[0], pA[aE] };
    v2f bv = { pB[0], pB[bE] };
    pA += 4*aE; pB += 4*bE;
    acc = __builtin_amdgcn_wmma_f32_16x16x4_f32(false, av, false, bv,
                                                (short)0, acc, false, false);
  }
  if (K4 < g.K) {                          // K tail (zero-padded)
    int ka = K4 + 2*half;
    float a0 = 0.f, a1 = 0.f, b0 = 0.f, b1 = 0.f;
    if (ka     < g.K) { a0 = pA[0];  b0 = pB[0];  }
    if (ka + 1 < g.K) { a1 = pA[aE]; b1 = pB[bE]; }
    v2f av = { a0, a1 }, bv = { b0, b1 };
    acc = __builtin_amdgcn_wmma_f32_16x16x4_f32(false, av, false, bv,
                                                (short)0, acc, false, false);
  }
  if (!bnv) return;
  const float* Add = g.Add;
  if (Add) Add += (long)i0*g.aS0 + (long)i1*g.aS1 + (long)i2*g.aS2;
  float* C = g.C + (long)i0*g.sC0 + (long)i1*g.sC1 + (long)i2*g.sC2;
  for (int r = 0; r < 8; ++r) {
    int m = m0 + half*8 + r;   // C layout: VGPR r -> row r (lanes 0-15) / r+8 (16-31)
    if (m >= g.M) break;
    C[(long)m*g.ldC + bn] = ep_apply(g, acc[r], bn, Add, (long)m);
  }
}

// ---- 32x32 tile per wave, 2x2 register blocking (heavy GEMMs, M,N >= 32) ----
// 8 loads -> 4 WMMAs per K-step: 2x the flop/byte of the 16x16 version.
__global__ __launch_bounds__(256) void wmma_gemm_f32_b2(GemmDesc g) {
  long wave = (long)blockIdx.x * 8 + (threadIdx.x >> 5);
  if (wave >= g.totalTiles) return;
  int lane = threadIdx.x & 31;
  int half = lane >> 4, ln = lane & 15;
  long t = wave;
  int jn = (int)(t % g.tn); t /= g.tn;
  int im = (int)(t % g.tm); t /= g.tm;
  int i2 = (int)(t % g.b2); t /= g.b2;
  int i1 = (int)(t % g.b1); t /= g.b1;
  int i0 = (int)t;
  const float* A  = g.A  + (long)i0*g.sA0 + (long)i1*g.sA1 + (long)i2*g.sA2;
  const float* Bp = g.Bm + (long)i0*g.sB0 + (long)i1*g.sB1 + (long)i2*g.sB2;
  int m0 = im*32, n0 = jn*32;
  int amc0 = m0 + ln;      if (amc0 > g.M - 1) amc0 = g.M - 1;
  int amc1 = m0 + 16 + ln; if (amc1 > g.M - 1) amc1 = g.M - 1;
  int bnc0 = n0 + ln;      if (bnc0 > g.N - 1) bnc0 = g.N - 1;
  int bnc1 = n0 + 16 + ln; if (bnc1 > g.N - 1) bnc1 = g.N - 1;
  const long aE = g.transA ? g.ldA : 1;
  const long bE = g.transB ? 1 : g.ldB;
  const long hA = (long)(2*half)*aE, hB = (long)(2*half)*bE;
  const float* pA0 = (g.transA ? (A + amc0) : (A + (long)amc0*g.ldA)) + hA;
  const float* pA1 = (g.transA ? (A + amc1) : (A + (long)amc1*g.ldA)) + hA;
  const float* pB0 = (g.transB ? (Bp + (long)bnc0*g.ldB) : (Bp + bnc0)) + hB;
  const float* pB1 = (g.transB ? (Bp + (long)bnc1*g.ldB) : (Bp + bnc1)) + hB;

  v8f a00 = {0.f,0.f,0.f,0.f,0.f,0.f,0.f,0.f};
  v8f a01 = a00, a10 = a00, a11 = a00;
  const int K4 = g.K & ~3;
  #pragma unroll 2
  for (int k0 = 0; k0 < K4; k0 += 4) {
    v2f av0 = { pA0[0], pA0[aE] };
    v2f av1 = { pA1[0], pA1[aE] };
    v2f bv0 = { pB0[0], pB0[bE] };
    v2f bv1 = { pB1[0], pB1[bE] };
    pA0 += 4*aE; pA1 += 4*aE; pB0 += 4*bE; pB1 += 4*bE;
    a00 = __builtin_amdgcn_wmma_f32_16x16x4_f32(false, av0, false, bv0, (short)0, a00, false, false);
    a01 = __builtin_amdgcn_wmma_f32_16x16x4_f32(false, av0, false, bv1, (short)0, a01, false, false);
    a10 = __builtin_amdgcn_wmma_f32_16x16x4_f32(false, av1, false, bv0, (short)0, a10, false, false);
    a11 = __builtin_amdgcn_wmma_f32_16x16x4_f32(false, av1, false, bv1, (short)0, a11, false, false);
  }
  if (K4 < g.K) {                          // K tail (zero-padded)
    int ka = K4 + 2*half;
    float x0 = 0.f, x1 = 0.f, y0 = 0.f, y1 = 0.f, u0 = 0.f, u1 = 0.f, w0 = 0.f, w1 = 0.f;
    if (ka     < g.K) { x0 = pA0[0];  y0 = pA1[0];  u0 = pB0[0];  w0 = pB1[0];  }
    if (ka + 1 < g.K) { x1 = pA0[aE]; y1 = pA1[aE]; u1 = pB0[bE]; w1 = pB1[bE]; }
    v2f av0 = { x0, x1 }, av1 = { y0, y1 }, bv0 = { u0, u1 }, bv1 = { w0, w1 };
    a00 = __builtin_amdgcn_wmma_f32_16x16x4_f32(false, av0, false, bv0, (short)0, a00, false, false);
    a01 = __builtin_amdgcn_wmma_f32_16x16x4_f32(false, av0, false, bv1, (short)0, a01, false, false);
    a10 = __builtin_amdgcn_wmma_f32_16x16x4_f32(false, av1, false, bv0, (short)0, a10, false, false);
    a11 = __builtin_amdgcn_wmma_f32_16x16x4_f32(false, av1, false, bv1, (short)0, a11, false, false);
  }
  const float* Add = g.Add;
  if (Add) Add += (long)i0*g.aS0 + (long)i1*g.aS1 + (long)i2*g.aS2;
  float* C = g.C + (long)i0*g.sC0 + (long)i1*g.sC1 + (long)i2*g.sC2;
  int nA = n0 + ln, nB = n0 + 16 + ln;
  bool nAv = nA < g.N, nBv = nB < g.N;
  for (int mi = 0; mi < 2; ++mi) {
    v8f* accL = mi ? &a10 : &a00;
    v8f* accR = mi ? &a11 : &a01;
    for (int r = 0; r < 8; ++r) {
      int m = m0 + mi*16 + half*8 + r;
      if (m >= g.M) break;
      long row = (long)m*g.ldC;
      if (nAv) C[row + nA] = ep_apply(g, (*accL)[r], nA, Add, (long)m);
      if (nBv) C[row + nB] = ep_apply(g, (*accR)[r], nB, Add, (long)m);
    }
  }
}

// ---------------- elementwise / reduction kernels ----------------
__global__ void k_norm_input(const float* x, float* z, float* zm, float* zs) {
  __shared__ float sh1[256], sh2[256];
  __shared__ float mS, sdS;
  int bc = blockIdx.x; int b = bc / CIN_, c = bc % CIN_;
  int tid = threadIdx.x;
  float s = 0.f, s2 = 0.f;
  for (int i = tid; i < S_; i += 256) {
    float v = x[((long)b*S_ + i)*CIN_ + c];
    s += v; s2 += v*v;
  }
  sh1[tid] = s; sh2[tid] = s2; __syncthreads();
  for (int o = 128; o > 0; o >>= 1) {
    if (tid < o) { sh1[tid] += sh1[tid+o]; sh2[tid] += sh2[tid+o]; }
    __syncthreads();
  }
  if (tid == 0) {
    float m = sh1[0] / (float)S_;
    float var = (sh2[0] - (float)S_*m*m) / (float)(S_ - 1);   // ddof=1
    float sd = sqrtf(fmaxf(var, 0.f));
    zm[bc] = m; zs[bc] = sd; mS = m; sdS = sd;
  }
  __syncthreads();
  float m = mS, inv = 1.f / (sdS + 1e-4f);
  for (int i = tid; i < S_; i += 256)
    z[(long)bc*S_ + i] = (x[((long)b*S_ + i)*CIN_ + c] - m) * inv;
}

__global__ void k_init_E(float* E) {
  int idx = blockIdx.x*blockDim.x + threadIdx.x;
  if (idx >= EMA_*EMA_) return;
  int i = idx / EMA_, j = idx % EMA_;
  float v = 0.f;
  if (i < TTN_ && j <= i)
    v = (j == 0) ? powf(0.7f, (float)i) : 0.3f * powf(0.7f, (float)(i - j));
  E[idx] = v;
}

__global__ void k_fill_cls(const float* cls, float* inp, long n) {
  long idx = (long)blockIdx.x*blockDim.x + threadIdx.x;
  if (idx >= n) return;
  int d = idx & (D_-1); long bc = idx >> 7;
  inp[bc*(long)TTN_*D_ + d] = cls[d];
}

__global__ void k_softmax(float* x, int W, long rows) {
  long r = (long)blockIdx.x*blockDim.x + threadIdx.x;
  if (r >= rows) return;
  float* p = x + r*(long)W;
  float mx = -1e30f;
  for (int w = 0; w < W; ++w) mx = fmaxf(mx, p[w]);
  float s = 0.f;
  for (int w = 0; w < W; ++w) { float e = __expf(p[w] - mx); p[w] = e; s += e; }
  float inv = 1.f / s;
  for (int w = 0; w < W; ++w) p[w] *= inv;
}

// merge: (BN, NH, H, HD) -> (BN, H, D) with cout = (m*4+g)*16+d2 permutation
__global__ void k_merge(const float* src, float* dst, int H, long total) {
  long idx = (long)blockIdx.x*blockDim.x + threadIdx.x;
  if (idx >= total) return;
  long t = idx;
  int cout = (int)(t % D_); t /= D_;
  int h2   = (int)(t % H);  t /= H;
  long bnb = t;
  int d2 = cout & 15, gg = (cout >> 4) & 3, mm = cout >> 6;
  long flat = (long)gg*(32L*H) + (long)h2*32 + mm*16 + d2;
  int nh = (int)(flat / (16L*H));
  long rem = flat % (16L*H);
  int hs = (int)(rem >> 4), d = (int)(rem & 15);
  dst[idx] = src[(((bnb*NH_ + nh)*(long)H + hs) << 4) + d];
}

__global__ void k_zero(float* p, int n) { if ((int)threadIdx.x < n) p[threadIdx.x] = 0.f; }

__global__ void k_bn_stats(const float* x, long rows, float* stats) {
  int ch = threadIdx.x & 127;
  float s = 0.f, s2 = 0.f;
  for (long r = blockIdx.x; r < rows; r += gridDim.x) {
    float v = x[r*D_ + ch]; s += v; s2 += v*v;
  }
  atomicAdd(&stats[ch], s); atomicAdd(&stats[D_ + ch], s2);
}

__global__ void k_bn_apply(const float* x, const float* stats, long rows, float invN,
                           float* dst, long r1, long r2, long ds0, long ds1, long ds2) {
  long idx = (long)blockIdx.x*blockDim.x + threadIdx.x;
  if (idx >= rows*D_) return;
  int ch = (int)(idx & (D_-1)); long r = idx >> 7;
  float m = stats[ch]*invN;
  float var = stats[D_+ch]*invN - m*m;
  float y = (x[idx] - m) * rsqrtf(var + 1e-5f);
  long i2 = r % r2; long q = r / r2; long i1 = q % r1; long i0 = q / r1;
  dst[i0*ds0 + i1*ds1 + i2*ds2 + ch] = y;
}

__global__ void k_add_flat(const float* a, const float* b, float* dst, long n) {
  long idx = (long)blockIdx.x*blockDim.x + threadIdx.x;
  if (idx < n) dst[idx] = a[idx] + b[idx];
}

// dst_flat = a_flat + strided view of b (rows decomposed (i0,i1,i2) over (.,r1,r2))
__global__ void k_add_sv(const float* a, const float* b, float* dst, long rows,
                         long r1, long r2, long bs0, long bs1, long bs2) {
  long idx = (long)blockIdx.x*blockDim.x + threadIdx.x;
  if (idx >= rows*D_) return;
  int ch = (int)(idx & (D_-1)); long r = idx >> 7;
  long i2 = r % r2; long q = r / r2; long i1 = q % r1; long i0 = q / r1;
  dst[idx] = a[idx] + b[i0*bs0 + i1*bs1 + i2*bs2 + ch];
}

__global__ void k_scale_out(const float* zout, const float* zm, const float* zs, float* out) {
  long idx = (long)blockIdx.x*blockDim.x + threadIdx.x;
  if (idx >= (long)B_*PRED_*CIN_) return;
  int c = (int)(idx % CIN_); long q = idx / CIN_;
  int p = (int)(q % PRED_); int b = (int)(q / PRED_);
  int bc = b*CIN_ + c;
  out[idx] = zout[(long)bc*PRED_ + p] * (zs[bc] + 1e-4f) + zm[bc];
}

// ---------------- host-side plumbing ----------------
static inline unsigned grid256(long n) { return (unsigned)((n + 255) / 256); }

static inline GemmDesc GD() {
  GemmDesc g{};                       // aggregate zero-init (host/device neutral)
  g.alpha = 1.f; g.b0 = g.b1 = g.b2 = 1;
  return g;
}

static inline void launch_gemm(hipStream_t st, GemmDesc g) {
  if (g.M >= 32 && g.N >= 32) {       // 2x2 register-blocked path (2x flop/byte)
    g.tm = (g.M + 31) >> 5; g.tn = (g.N + 31) >> 5;
    g.totalTiles = (long)g.b0 * g.b1 * g.b2 * g.tm * g.tn;
    unsigned blocks = (unsigned)((g.totalTiles + 7) / 8);
    wmma_gemm_f32_b2<<<blocks, 256, 0, st>>>(g);
  } else {
    g.tm = (g.M + 15) >> 4; g.tn = (g.N + 15) >> 4;
    g.totalTiles = (long)g.b0 * g.b1 * g.b2 * g.tm * g.tn;
    unsigned blocks = (unsigned)((g.totalTiles + 7) / 8);  // 8 waves / 256-thread block
    wmma_gemm_f32<<<blocks, 256, 0, st>>>(g);
  }
}

struct AttnW {
  const float *qkv_w,*qkv_b,*dpk_w,*dpk_b,*dpv_w,*dpv_b,
              *ff1_w1,*ff1_b1,*ff1_w2,*ff1_b2,*ff2_w1,*ff2_b1,*ff2_w2,*ff2_b2;
};
struct LayerW { AttnW chan, tok; const float *mlp_w, *mlp_b; };
struct Pools {
  float *E,*zm,*zs,*stats,*z,*inp,*o1,*o2,*qkv,*t0,*t1,*t2,*t3,*t4,*t5,*ka,*va;
};

// workspace layout (floats)
static constexpr long SZ_INP = (long)B_*CIN_*TTN_*D_;        // 14,791,680
static constexpr long SZ_QKV = 3*SZ_INP;                     // 44,375,040
static constexpr long SZ_SC  = (long)B_*CIN_*NH_*TTN_*TTN_;  // 83,203,200 (tok scores; also s / ff-hidden)
static constexpr long SZ_T5  = (long)B_*CIN_*NH_*HD_*HD_;    //  2,629,632
static constexpr long SZ_KV  = (long)B_*TTN_*NH_*DP_*HD_;    //    368,640
static constexpr long OFF_E   = 0;
static constexpr long OFF_ZM  = OFF_E   + (long)EMA_*EMA_;
static constexpr long OFF_ZS  = OFF_ZM  + (long)B_*CIN_;
static constexpr long OFF_ST  = OFF_ZS  + (long)B_*CIN_;
static constexpr long OFF_Z   = OFF_ST  + 256;
static constexpr long OFF_INP = OFF_Z   + (long)B_*CIN_*S_;
static constexpr long OFF_O1  = OFF_INP + SZ_INP;
static constexpr long OFF_O2  = OFF_O1  + SZ_INP;
static constexpr long OFF_QKV = OFF_O2  + SZ_INP;
static constexpr long OFF_T0  = OFF_QKV + SZ_QKV;
static constexpr long OFF_T1  = OFF_T0  + SZ_INP;
static constexpr long OFF_T3  = OFF_T1  + SZ_INP;
static constexpr long OFF_T4  = OFF_T3  + SZ_INP;
static constexpr long OFF_T5  = OFF_T4  + SZ_INP;
static constexpr long OFF_KA  = OFF_T5  + SZ_T5;
static constexpr long OFF_VA  = OFF_KA  + SZ_KV;
static constexpr long OFF_T2  = OFF_VA  + SZ_KV;

static void bn(hipStream_t st, Pools& P, const float* src, long rows,
               float* dst, long r1, long r2, long d0, long d1, long d2) {
  k_zero<<<1, 256, 0, st>>>(P.stats, 256);
  k_bn_stats<<<512, 128, 0, st>>>(src, rows, P.stats);
  k_bn_apply<<<grid256(rows*D_), 256, 0, st>>>(src, P.stats, rows,
                                               1.0f/(float)rows, dst, r1, r2, d0, d1, d2);
}

// src/dst viewed as [b, n, h, c] with element strides (sb, sn, sh, 1).
static void attention(hipStream_t st, Pools& P, const AttnW& w,
                      const float* src, long sb, long sn, long sh,
                      float* dst, long ob, long on, long oh,
                      int n, int H, bool over_hidden) {
  const long rows = (long)B_*n*H;
  const long q0 = (long)n*H*3*D_, q1 = (long)H*3*D_, q2 = HD_;   // (b,n,nh) strides into qkv
  // 1) qkv = src @ qkv_w + b            rows (b,n,h) x 384
  { GemmDesc g = GD();
    g.A = src; g.ldA = sh; g.sA0 = sb; g.sA1 = sn;
    g.Bm = w.qkv_w; g.ldB = 3*D_;
    g.bias = w.qkv_b;
    g.C = P.qkv; g.ldC = 3*D_; g.sC0 = q0; g.sC1 = q1;
    g.M = H; g.N = 3*D_; g.K = D_; g.b0 = B_; g.b1 = n;
    launch_gemm(st, g); }
  const float* q  = P.qkv;
  const float* kk = P.qkv + D_;
  const float* vv = P.qkv + 2*D_;

  int F; const float *kf, *vf; long kld, kf0, kf1, kf2, vld, vf0, vf1, vf2;
  if (over_hidden) {
    // dyn_proj for k and v: s = softmax(t @ w + b); t_ = s^T @ t   (DPxHD)
    float* s = P.t2;
    const long s0 = (long)n*NH_*H*DP_, s1 = (long)NH_*H*DP_, s2 = (long)H*DP_;
    const long kb0 = (long)n*NH_*DP_*HD_, kb1 = (long)NH_*DP_*HD_, kb2 = (long)DP_*HD_;
    const float* tsrc[2] = { kk, vv };
    const float* wdp[2]  = { w.dpk_w, w.dpv_w };
    const float* bdp[2]  = { w.dpk_b, w.dpv_b };
    float* outp[2] = { P.ka, P.va };
    for (int i = 0; i < 2; ++i) {
      { GemmDesc g = GD();
        g.A = tsrc[i]; g.ldA = 3*D_; g.sA0 = q0; g.sA1 = q1; g.sA2 = q2;
        g.Bm = wdp[i]; g.ldB = DP_; g.bias = bdp[i];
        g.C = s; g.ldC = DP_; g.sC0 = s0; g.sC1 = s1; g.sC2 = s2;
        g.M = H; g.N = DP_; g.K = HD_; g.b0 = B_; g.b1 = n; g.b2 = NH_;
        launch_gemm(st, g); }
      k_softmax<<<grid256((long)B_*n*NH_*H), 256, 0, st>>>(s, DP_, (long)B_*n*NH_*H);
      { GemmDesc g = GD();                       // t_ = s^T @ t : M=DP, K=H, N=HD
        g.A = s; g.ldA = DP_; g.sA0 = s0; g.sA1 = s1; g.sA2 = s2; g.transA = 1;
        g.Bm = tsrc[i]; g.ldB = 3*D_; g.sB0 = q0; g.sB1 = q1; g.sB2 = q2;
        g.C = outp[i]; g.ldC = HD_; g.sC0 = kb0; g.sC1 = kb1; g.sC2 = kb2;
        g.M = DP_; g.N = HD_; g.K = H; g.b0 = B_; g.b1 = n; g.b2 = NH_;
        launch_gemm(st, g); }
    }
    F = DP_;
    kf = P.ka; kld = HD_; kf0 = kb0; kf1 = kb1; kf2 = kb2;
    vf = P.va; vld = HD_; vf0 = kb0; vf1 = kb1; vf2 = kb2;
  } else {
    F = H;
    kf = kk; kld = 3*D_; kf0 = q0; kf1 = q1; kf2 = q2;
    vf = vv; vld = 3*D_; vf0 = q0; vf1 = q1; vf2 = q2;
  }
  const long e0 = (long)n*NH_*H*HD_, e1 = (long)NH_*H*HD_, e2 = (long)H*HD_;
  // 2) emaq = E[:H,:H] @ q -> t0
  { GemmDesc g = GD();
    g.A = P.E; g.ldA = EMA_;
    g.Bm = q; g.ldB = 3*D_; g.sB0 = q0; g.sB1 = q1; g.sB2 = q2;
    g.C = P.t0; g.ldC = HD_; g.sC0 = e0; g.sC1 = e1; g.sC2 = e2;
    g.M = H; g.N = HD_; g.K = H; g.b0 = B_; g.b1 = n; g.b2 = NH_;
    launch_gemm(st, g); }
  // 3) emak = E[:F,:F] @ k_ -> t1
  const long g0 = (long)n*NH_*F*HD_, g1 = (long)NH_*F*HD_, g2 = (long)F*HD_;
  { GemmDesc g = GD();
    g.A = P.E; g.ldA = EMA_;
    g.Bm = kf; g.ldB = kld; g.sB0 = kf0; g.sB1 = kf1; g.sB2 = kf2;
    g.C = P.t1; g.ldC = HD_; g.sC0 = g0; g.sC1 = g1; g.sC2 = g2;
    g.M = F; g.N = HD_; g.K = F; g.b0 = B_; g.b1 = n; g.b2 = NH_;
    launch_gemm(st, g); }
  // 4) score_t = emaq @ emak^T * sqrt(HD) -> t2 ; softmax over F
  const long sc0 = (long)n*NH_*H*F, sc1 = (long)NH_*H*F, sc2 = (long)H*F;
  { GemmDesc g = GD();
    g.A = P.t0; g.ldA = HD_; g.sA0 = e0; g.sA1 = e1; g.sA2 = e2;
    g.Bm = P.t1; g.ldB = HD_; g.sB0 = g0; g.sB1 = g1; g.sB2 = g2; g.transB = 1;
    g.C = P.t2; g.ldC = F; g.sC0 = sc0; g.sC1 = sc1; g.sC2 = sc2;
    g.M = H; g.N = F; g.K = HD_; g.b0 = B_; g.b1 = n; g.b2 = NH_; g.alpha = 4.0f;
    launch_gemm(st, g); }
  k_softmax<<<grid256((long)B_*n*NH_*H), 256, 0, st>>>(P.t2, F, (long)B_*n*NH_*H);
  // 5) out_t = P @ v_ -> t3
  { GemmDesc g = GD();
    g.A = P.t2; g.ldA = F; g.sA0 = sc0; g.sA1 = sc1; g.sA2 = sc2;
    g.Bm = vf; g.ldB = vld; g.sB0 = vf0; g.sB1 = vf1; g.sB2 = vf2;
    g.C = P.t3; g.ldC = HD_; g.sC0 = e0; g.sC1 = e1; g.sC2 = e2;
    g.M = H; g.N = HD_; g.K = F; g.b0 = B_; g.b1 = n; g.b2 = NH_;
    launch_gemm(st, g); }
  // 6) score_h = q^T @ k * sqrt(H) -> t5 ; softmax over 16
  const long h0 = (long)n*NH_*HD_*HD_, h1 = (long)NH_*HD_*HD_, h2s = (long)HD_*HD_;
  { GemmDesc g = GD();
    g.A = q; g.ldA = 3*D_; g.sA0 = q0; g.sA1 = q1; g.sA2 = q2; g.transA = 1;
    g.Bm = kk; g.ldB = 3*D_; g.sB0 = q0; g.sB1 = q1; g.sB2 = q2;
    g.C = P.t5; g.ldC = HD_; g.sC0 = h0; g.sC1 = h1; g.sC2 = h2s;
    g.M = HD_; g.N = HD_; g.K = H; g.b0 = B_; g.b1 = n; g.b2 = NH_;
    g.alpha = sqrtf((float)H);
    launch_gemm(st, g); }
  k_softmax<<<grid256((long)B_*n*NH_*HD_), 256, 0, st>>>(P.t5, HD_, (long)B_*n*NH_*HD_);
  // 7) out_h = v @ P^T -> t4
  { GemmDesc g = GD();
    g.A = vv; g.ldA = 3*D_; g.sA0 = q0; g.sA1 = q1; g.sA2 = q2;
    g.Bm = P.t5; g.ldB = HD_; g.sB0 = h0; g.sB1 = h1; g.sB2 = h2s; g.transB = 1;
    g.C = P.t4; g.ldC = HD_; g.sC0 = e0; g.sC1 = e1; g.sC2 = e2;
    g.M = H; g.N = HD_; g.K = HD_; g.b0 = B_; g.b1 = n; g.b2 = NH_;
    launch_gemm(st, g); }
  // 8) merge + bn
  long mtotal = (long)B_*n*H*D_;
  k_merge<<<grid256(mtotal), 256, 0, st>>>(P.t3, P.t0, H, mtotal);
  k_merge<<<grid256(mtotal), 256, 0, st>>>(P.t4, P.t1, H, mtotal);
  bn(st, P, P.t0, rows, P.t0, 1, rows, 0, 0, (long)D_);
  bn(st, P, P.t1, rows, P.t1, 1, rows, 0, 0, (long)D_);
  // 9) src2 = ff1(o1m) + ff2(o2m) -> t3 (second FF accumulates via Add=C)
  const float* fin[2] = { P.t0, P.t1 };
  const float* w1[2]  = { w.ff1_w1, w.ff2_w1 }; const float* b1[2] = { w.ff1_b1, w.ff2_b1 };
  const float* w2[2]  = { w.ff1_w2, w.ff2_w2 }; const float* b2[2] = { w.ff1_b2, w.ff2_b2 };
  for (int i = 0; i < 2; ++i) {
    { GemmDesc g = GD();                // hidden = gelu(x @ w1 + b1) -> t2
      g.A = fin[i]; g.ldA = D_;
      g.Bm = w1[i]; g.ldB = DFF_; g.bias = b1[i]; g.gelu = 1;
      g.C = P.t2; g.ldC = DFF_;
      g.M = (int)rows; g.N = DFF_; g.K = D_;
      launch_gemm(st, g); }
    { GemmDesc g = GD();                // t3 (+)= hidden @ w2 + b2
      g.A = P.t2; g.ldA = DFF_;
      g.Bm = w2[i]; g.ldB = D_; g.bias = b2[i];
      if (i == 1) { g.Add = P.t3; g.ldAdd = D_; }
      g.C = P.t3; g.ldC = D_;
      g.M = (int)rows; g.N = D_; g.K = DFF_;
      launch_gemm(st, g); }
  }
  // 10) out = bn(src + src2) written into strided dst
  k_add_sv<<<grid256(rows*D_), 256, 0, st>>>(P.t3, src, P.t4, rows, (long)n, (long)H, sb, sn, sh);
  bn(st, P, P.t4, rows, dst, (long)n, (long)H, ob, on, oh);
}

extern "C" void kernel_launch(void* const* d_in, const int* in_sizes, int n_in,
                              void* d_out, int out_size, void* d_ws, size_t ws_size,
                              hipStream_t stream) {
  (void)in_sizes; (void)n_in; (void)out_size; (void)ws_size;
  int idx = 0;
  auto nxt = [&]() { return (const float*)d_in[idx++]; };
  const float* x    = nxt();
  const float* Wp   = nxt();
  const float* bp   = nxt();
  const float* pos  = nxt();
  const float* cls  = nxt();
  const float* Wout = nxt();
  const float* bout = nxt();
  LayerW L[EL_];
  for (int l = 0; l < EL_; ++l) {
    auto loadA = [&](AttnW& a) {
      a.qkv_w = nxt(); a.qkv_b = nxt();
      a.dpk_w = nxt(); a.dpk_b = nxt(); a.dpv_w = nxt(); a.dpv_b = nxt();
      a.ff1_w1 = nxt(); a.ff1_b1 = nxt(); a.ff1_w2 = nxt(); a.ff1_b2 = nxt();
      a.ff2_w1 = nxt(); a.ff2_b1 = nxt(); a.ff2_w2 = nxt(); a.ff2_b2 = nxt();
    };
    loadA(L[l].chan); loadA(L[l].tok);
    L[l].mlp_w = nxt(); L[l].mlp_b = nxt();
  }
  float* wsf = (float*)d_ws;
  Pools P;
  P.E = wsf + OFF_E; P.zm = wsf + OFF_ZM; P.zs = wsf + OFF_ZS; P.stats = wsf + OFF_ST;
  P.z = wsf + OFF_Z; P.inp = wsf + OFF_INP; P.o1 = wsf + OFF_O1; P.o2 = wsf + OFF_O2;
  P.qkv = wsf + OFF_QKV; P.t0 = wsf + OFF_T0; P.t1 = wsf + OFF_T1; P.t2 = wsf + OFF_T2;
  P.t3 = wsf + OFF_T3; P.t4 = wsf + OFF_T4; P.t5 = wsf + OFF_T5;
  P.ka = wsf + OFF_KA; P.va = wsf + OFF_VA;

  const long CTD = (long)CIN_*TTN_*D_;   // (B,C,T,D) batch stride
  const long TTD = (long)TTN_*D_;

  // Stage 0: per-(b,c) z-norm (ddof=1), EMA matrix, cls token
  k_norm_input<<<B_*CIN_, 256, 0, stream>>>(x, P.z, P.zm, P.zs);
  k_init_E<<<grid256((long)EMA_*EMA_), 256, 0, stream>>>(P.E);
  k_fill_cls<<<grid256((long)B_*CIN_*D_), 256, 0, stream>>>(cls, P.inp, (long)B_*CIN_*D_);

  // Stage 1: patch embedding (overlapping windows: lda = STRIDE), +bp +pos
  { GemmDesc g = GD();
    g.A = P.z; g.ldA = STRIDE_; g.sA0 = (long)CIN_*S_; g.sA1 = (long)S_;
    g.Bm = Wp; g.ldB = D_; g.bias = bp;
    g.Add = pos; g.ldAdd = D_;                      // pos[p,d], batch-invariant
    g.C = P.inp + D_; g.ldC = D_; g.sC0 = CTD; g.sC1 = TTD;  // tokens 1..PN
    g.M = PN_; g.N = D_; g.K = PATCH_; g.b0 = B_; g.b1 = CIN_;
    launch_gemm(stream, g); }

  // Stage 2: encoder layers
  const long rowsL = (long)B_*CIN_*TTN_;
  for (int l = 0; l < EL_; ++l) {
    // channel attention: src view (b, t, c, d) of inp ; output back into (B,C,T,D) o1
    attention(stream, P, L[l].chan,
              P.inp, CTD, (long)D_, TTD,
              P.o1,  CTD, (long)D_, TTD,
              TTN_, CIN_, true);
    // token attention: src o1 (b, c, t, d) ; output o2 contiguous
    attention(stream, P, L[l].tok,
              P.o1, CTD, TTD, (long)D_,
              P.o2, CTD, TTD, (long)D_,
              CIN_, TTN_, false);
    // out = (o1+o2) @ mlp_w + mlp_b + inp ; inp = bn(out)
    k_add_flat<<<grid256(SZ_INP), 256, 0, stream>>>(P.o1, P.o2, P.t0, SZ_INP);
    { GemmDesc g = GD();
      g.A = P.t0; g.ldA = D_;
      g.Bm = L[l].mlp_w; g.ldB = D_; g.bias = L[l].mlp_b;
      g.Add = P.inp; g.ldAdd = D_;
      g.C = P.t3; g.ldC = D_;
      g.M = (int)rowsL; g.N = D_; g.K = D_;
      launch_gemm(stream, g); }
    bn(stream, P, P.t3, rowsL, P.inp, 1, rowsL, 0, 0, (long)D_);
  }

  // Stage 3: head  zout = inp.reshape(B*CIN, TTN*D) @ Wout + bout ; rescale + transpose
  { GemmDesc g = GD();
    g.A = P.inp; g.ldA = TTD;
    g.Bm = Wout; g.ldB = PRED_; g.bias = bout;
    g.C = P.t0; g.ldC = PRED_;
    g.M = B_*CIN_; g.N = PRED_; g.K = (int)TTD;
    launch_gemm(stream, g); }
  k_scale_out<<<grid256((long)B_*PRED_*CIN_), 256, 0, stream>>>(P.t0, P.zm, P.zs, (float*)d_out);
}